// ELinear_9766755631223
// MI455X (gfx1250) — compile-verified
//
#include <hip/hip_runtime.h>

typedef __attribute__((ext_vector_type(2))) float v2f;
typedef __attribute__((ext_vector_type(8))) float v8f;

// Problem constants: x (4096,128,64), w (128,64,64), bias (128,64), all fp32
#define D_IN      64
#define D_OUT     64
#define N_TOK     128
#define ROWS_PER_BLOCK 128      // 8 waves * 16 rows

// LDS holds W_t pre-swizzled into WMMA B-fragment layout:
//   fragment f = ks*4 + nt  (ks = K/4 step 0..15, nt = 16-col tile 0..3)
//   lane L of fragment f holds float2 { W[kb][c], W[kb+1][c] }
//     kb = ks*4 + (L>>4)*2,  c = nt*16 + (L&15)
// Flat: sW[(f*32 + L)*2 + {0,1}]  -> one ds_load_b64 per fragment, bank-conflict-free.

__global__ __launch_bounds__(256) void
elinear_wmma_f32(const float* __restrict__ x,
                 const float* __restrict__ w,
                 const float* __restrict__ bias,
                 float* __restrict__ out)
{
    const int t     = blockIdx.y;                 // token
    const int bBase = blockIdx.x * ROWS_PER_BLOCK;

    __shared__ __align__(16) float sW[D_IN * D_OUT];  // 16 KB, swizzled

    const int tid  = threadIdx.x;                 // 0..255
    const int lane = tid & 31;
    const int wave = tid >> 5;                    // 0..7

    // ---- Stage W_t (16 KB) into LDS in B-fragment order ----
    {
        const float* wt = w + (size_t)t * (D_IN * D_OUT);
        #pragma unroll
        for (int i = 0; i < 4; ++i) {
            const int g  = (tid + i * 256) * 4;       // flat float idx, row-aligned x4
            const int r  = g >> 6;                    // K row 0..63
            const int c0 = g & 63;                    // col (multiple of 4)
            const float4 v = *(const float4*)&wt[g];
            const int part   = r & 1;                 // .x or .y of the pair
            const int laneHi = ((r >> 1) & 1) << 4;   // lane group select
            const int f      = (r >> 2) * 4 + (c0 >> 4);
            const int lane0  = laneHi + (c0 & 15);
            const float e[4] = {v.x, v.y, v.z, v.w};
            #pragma unroll
            for (int j = 0; j < 4; ++j)
                sW[(f * 32 + lane0 + j) * 2 + part] = e[j];
        }
    }
    __syncthreads();

    // ---- Per-wave 16x64 output strip ----
    const int n    = lane & 15;          // col within 16-wide tile / A row
    const int kOff = (lane >> 4) * 2;    // A K-pair select: 0 or 2
    const int rowBase = bBase + wave * 16;

    // Preload all A fragments: x[rowBase+n][t][k..k+1], 16 x global_load_b64
    const float* xrow = x + ((size_t)(rowBase + n) * N_TOK + t) * D_IN;
    v2f a[16];
    #pragma unroll
    for (int ks = 0; ks < 16; ++ks)
        a[ks] = *(const v2f*)&xrow[ks * 4 + kOff];

    // Init accumulators with bias (depends only on output column)
    v8f acc[4];
    #pragma unroll
    for (int nt = 0; nt < 4; ++nt) {
        const float bv = bias[(size_t)t * D_OUT + nt * 16 + n];
        #pragma unroll
        for (int r = 0; r < 8; ++r) acc[nt][r] = bv;
    }

    // K loop: 16 steps x 4 col tiles -> 64 (ds_load_b64 + v_wmma_f32_16x16x4_f32)
    #pragma unroll
    for (int ks = 0; ks < 16; ++ks) {
        #pragma unroll
        for (int nt = 0; nt < 4; ++nt) {
            const v2f b = *(const v2f*)&sW[((ks * 4 + nt) * 32 + lane) * 2];
            acc[nt] = __builtin_amdgcn_wmma_f32_16x16x4_f32(
                /*neg_a=*/false, a[ks], /*neg_b=*/false, b,
                /*c_mod=*/(short)0, acc[nt],
                /*reuse_a=*/false, /*reuse_b=*/false);
        }
    }

    // ---- Store: element r of acc[nt] -> row (lane>>4)*8 + r, col nt*16 + n ----
    const int mHi = (lane >> 4) * 8;
    #pragma unroll
    for (int r = 0; r < 8; ++r) {
        float* o = out + ((size_t)(rowBase + mHi + r) * N_TOK + t) * D_OUT;
        #pragma unroll
        for (int nt = 0; nt < 4; ++nt)
            o[nt * 16 + n] = acc[nt][r];
    }
}

extern "C" void kernel_launch(void* const* d_in, const int* in_sizes, int n_in,
                              void* d_out, int out_size, void* d_ws, size_t ws_size,
                              hipStream_t stream)
{
    const float* x    = (const float*)d_in[0];   // (4096,128,64)
    const float* w    = (const float*)d_in[1];   // (128,64,64)
    const float* bias = (const float*)d_in[2];   // (128,64)
    float* out        = (float*)d_out;           // (4096,128,64)

    const int batch = in_sizes[0] / (N_TOK * D_IN);   // 4096
    dim3 grid(batch / ROWS_PER_BLOCK, N_TOK);         // (32, 128)
    dim3 block(256);
    elinear_wmma_f32<<<grid, block, 0, stream>>>(x, w, bias, out);
}